// HierSoftmaxCrossEntropy_37555194036885
// MI455X (gfx1250) — compile-verified
//
#include <hip/hip_runtime.h>
#include <stdint.h>

// Hierarchical softmax cross-entropy, MI455X (gfx1250).
// One workgroup (256 threads = 8 waves, wave32) per batch row.
// Score row staged to LDS via CDNA5 async global->LDS loads (ASYNCcnt).

#define TB 256
#define MAX_NJ 8192   // num_nodes-1 <= 7999 by construction
#define MAX_NI 4096   // every internal node has >=2 children

// ---------------- prep kernels ----------------

__global__ void prep_zero_kernel(int* gstart, int* gcount, float* gprior,
                                 float* out, int nj) {
    int j = blockIdx.x * blockDim.x + threadIdx.x;
    if (j < nj) {
        gstart[j] = 0x7FFFFFFF;
        gcount[j] = 0;
        gprior[j] = 0.0f;
    }
    if (j == 0) out[0] = 0.0f;
}

__global__ void prep_groups_kernel(const int* __restrict__ flat_index,
                                   const float* __restrict__ prior,
                                   const int* __restrict__ p_maxc,
                                   int* __restrict__ gid,
                                   int* __restrict__ gstart,
                                   int* __restrict__ gcount,
                                   float* __restrict__ gprior, int nj) {
    int j = blockIdx.x * blockDim.x + threadIdx.x;
    if (j >= nj) return;
    const int mc = *p_maxc;
    const int g  = flat_index[j] / mc;      // parent internal-node id of node j+1
    gid[j] = g;
    atomicMin(&gstart[g], j);               // children of g are contiguous in j
    atomicAdd(&gcount[g], 1);
    atomicAdd(&gprior[g], prior[j + 1]);    // prior mass of g's children
}

// ---------------- main kernel ----------------

__global__ __launch_bounds__(TB) void hier_xent_kernel(
    const float* __restrict__ scores,   // [B, nj]
    const int*   __restrict__ labels,   // [B]
    const float* __restrict__ prior,    // [nj+1]
    const int*   __restrict__ gid,      // [nj]
    const int*   __restrict__ gstart,   // [ni]
    const int*   __restrict__ gcount,   // [ni]
    const float* __restrict__ gprior,   // [ni]
    const int*   __restrict__ p_ni,     // scalar num_internal (device)
    float*       __restrict__ out,      // [1]
    int nj, float inv_b) {

    __shared__ float s_scores[MAX_NJ];
    __shared__ float s_lse[MAX_NI];
    __shared__ float s_red[TB];

    const int tid = threadIdx.x;
    const int b   = blockIdx.x;
    const int ni  = *p_ni;

    // ---- stage this row's scores into LDS via async global->LDS copy ----
    // GVS mode: saddr = 64-bit row base (uniform), vaddr = per-lane byte offset,
    // vdst = per-lane LDS byte address.  Tracked by ASYNCcnt.
    {
        const uint64_t base = (uint64_t)(uintptr_t)scores
                            + (uint64_t)b * (uint64_t)nj * 4ull;
        for (int j = tid; j < nj; j += TB) {
            uint32_t lds_addr = (uint32_t)(uintptr_t)(&s_scores[j]);
            uint32_t voff     = (uint32_t)j * 4u;
            asm volatile("global_load_async_to_lds_b32 %0, %1, %2"
                         :
                         : "v"(lds_addr), "v"(voff), "s"(base)
                         : "memory");
        }
        asm volatile("s_wait_asynccnt 0" ::: "memory");
    }
    __syncthreads();

    // ---- per-group log-sum-exp (children of group i are s_scores[st..st+c)) ----
    for (int i = tid; i < ni; i += TB) {
        const int st = gstart[i];
        const int c  = gcount[i];
        float m = -3.402823466e38f;
        for (int k = 0; k < c; ++k) m = fmaxf(m, s_scores[st + k]);
        float ssum = 0.0f;
        for (int k = 0; k < c; ++k) ssum += __expf(s_scores[st + k] - m);
        s_lse[i] = m + __logf(ssum);
    }
    __syncthreads();

    // ---- prior term: -sum_j prior[j+1]*score[j] + sum_i gprior[i]*lse[i] ----
    float v = 0.0f;
    for (int j = tid; j < nj; j += TB) v -= prior[j + 1] * s_scores[j];
    for (int i = tid; i < ni; i += TB) v += gprior[i] * s_lse[i];

    s_red[tid] = v;
    __syncthreads();
    #pragma unroll
    for (int s = TB / 2; s > 0; s >>= 1) {
        if (tid < s) s_red[tid] += s_red[tid + s];
        __syncthreads();
    }

    // ---- path term: walk leaf -> root (parent(n) = gid[n-1]) ----
    if (tid == 0) {
        const float priorterm = s_red[0];
        int   node = ni + labels[b];        // leaf node id
        float path = 0.0f;
        while (node != 0) {
            const int j = node - 1;
            const int g = gid[j];
            path += s_scores[j] - s_lse[g]; // log_cond_p at this node
            node = g;
        }
        const float xent = 0.9f * (-path) + 0.1f * priorterm;
        atomicAdd(out, xent * inv_b);
    }
}

// ---------------- launcher ----------------

extern "C" void kernel_launch(void* const* d_in, const int* in_sizes, int n_in,
                              void* d_out, int out_size, void* d_ws, size_t ws_size,
                              hipStream_t stream) {
    (void)n_in; (void)out_size; (void)ws_size;

    const float* scores     = (const float*)d_in[0];
    const int*   labels     = (const int*)  d_in[1];
    // d_in[2] (matrix) is never read: path walk replaces the 65 MB row gather.
    const float* prior      = (const float*)d_in[3];
    const int*   flat_index = (const int*)  d_in[4];
    // d_in[5] (child_index) implicit: child_index[j] == j+1.
    const int*   p_ni       = (const int*)  d_in[6];
    const int*   p_mc       = (const int*)  d_in[7];

    const int B  = in_sizes[1];             // 2048
    int nj       = in_sizes[4];             // num_nodes - 1
    if (nj > MAX_NJ) nj = MAX_NJ;

    char* ws = (char*)d_ws;
    int*   gid    = (int*)ws;   ws += (size_t)nj * sizeof(int);
    int*   gstart = (int*)ws;   ws += (size_t)nj * sizeof(int);
    int*   gcount = (int*)ws;   ws += (size_t)nj * sizeof(int);
    float* gprior = (float*)ws;
    float* out    = (float*)d_out;

    const int gb = (nj + TB - 1) / TB;
    prep_zero_kernel  <<<gb, TB, 0, stream>>>(gstart, gcount, gprior, out, nj);
    prep_groups_kernel<<<gb, TB, 0, stream>>>(flat_index, prior, p_mc,
                                              gid, gstart, gcount, gprior, nj);
    hier_xent_kernel  <<<B,  TB, 0, stream>>>(scores, labels, prior,
                                              gid, gstart, gcount, gprior,
                                              p_ni, out, nj, 1.0f / (float)B);
}